// Mamba2Mixer_51453708206607
// MI455X (gfx1250) — compile-verified
//
#include <hip/hip_runtime.h>
#include <math.h>

// ---- Mamba2-1.3B-like config (must match reference) ----
#define B_        2
#define S_        1024
#define HID_      2048
#define INTER_    4096
#define NHEADS_   64
#define HDIM_     64
#define NSTATE_   64
#define KCONV_    4
#define CONV_DIM_ 4224
#define PROJ_     8384          // INTER + CONV_DIM + NHEADS
#define DT_OFF_   (INTER_ + CONV_DIM_)
#define BS_       (B_ * S_)
#define EPS_      1e-6f

typedef float v2f __attribute__((ext_vector_type(2)));
typedef float v4f __attribute__((ext_vector_type(4)));
typedef float v8f __attribute__((ext_vector_type(8)));
typedef int   v4i __attribute__((ext_vector_type(4)));

// ---------------------------------------------------------------------------
// CDNA5 async global->LDS copies (GLOBAL_LOAD_ASYNC_TO_LDS_*, ASYNCcnt).
// Prototype (from hipcc diagnostics): b128 takes (AS1 v4i*, AS3 v4i*, Ii, Ii),
// b32 takes (AS1 int*, AS3 int*, Ii, Ii). Guarded with a synchronous fallback.
// ---------------------------------------------------------------------------
#if defined(__has_builtin)
#if __has_builtin(__builtin_amdgcn_global_load_async_to_lds_b128)
#define HAVE_ASYNC_LDS 1
#endif
#endif
#ifndef HAVE_ASYNC_LDS
#define HAVE_ASYNC_LDS 0
#endif

#if HAVE_ASYNC_LDS && defined(__has_builtin)
#if __has_builtin(__builtin_amdgcn_s_wait_asynccnt)
#define HAVE_WAIT_ASYNC_BUILTIN 1
#endif
#endif
#ifndef HAVE_WAIT_ASYNC_BUILTIN
#define HAVE_WAIT_ASYNC_BUILTIN 0
#endif

typedef __attribute__((address_space(1))) v4i g_v4i;
typedef __attribute__((address_space(3))) v4i l_v4i;
typedef __attribute__((address_space(1))) int g_i32;
typedef __attribute__((address_space(3))) int l_i32;

__device__ __forceinline__ void cp_async16(const void* g, void* l) {
#if HAVE_ASYNC_LDS
    // generic LDS pointer: low 32 bits == AS3 offset (addrspacecast layout)
    __builtin_amdgcn_global_load_async_to_lds_b128(
        (g_v4i*)(unsigned long long)g,
        (l_v4i*)(unsigned int)(unsigned long long)l, 0, 0);
#else
    *(v4f*)l = *(const v4f*)g;
#endif
}

__device__ __forceinline__ void cp_async4(const void* g, void* l) {
#if HAVE_ASYNC_LDS && defined(__has_builtin) && __has_builtin(__builtin_amdgcn_global_load_async_to_lds_b32)
    __builtin_amdgcn_global_load_async_to_lds_b32(
        (g_i32*)(unsigned long long)g,
        (l_i32*)(unsigned int)(unsigned long long)l, 0, 0);
#else
    *(float*)l = *(const float*)g;
#endif
}

__device__ __forceinline__ void wait_async_lds() {
#if HAVE_WAIT_ASYNC_BUILTIN
    __builtin_amdgcn_s_wait_asynccnt(0);
#elif HAVE_ASYNC_LDS
    asm volatile("s_wait_asynccnt 0x0" ::: "memory");
#endif
}

// ---------------------------------------------------------------------------
// Fp32 GEMM via V_WMMA_F32_16X16X4_F32.
// C[M,N] = A[M,K] * B[K,N], all row-major fp32.
// Block tile 128x128, K-step 32, 256 threads = 8 waves (wave32),
// wave grid 2x4, each wave computes 64x32 (4x2 tiles of 16x16).
// Double-buffered LDS; tiles staged with async global->LDS copies.
// M must be a multiple of 128, K a multiple of 32; N is guarded.
// ---------------------------------------------------------------------------
__global__ __launch_bounds__(256) void gemm_f32_wmma(
    const float* __restrict__ A, const float* __restrict__ Bg,
    float* __restrict__ C, int M, int N, int K)
{
    __shared__ float As[2][128][36];   // pad 36: 16B-aligned, conflict-free frags
    __shared__ float Bs[2][32][132];

    const int tid   = threadIdx.x;
    const int wave  = tid >> 5;
    const int lane  = tid & 31;
    const int lhalf = lane >> 4;    // 0: lanes 0-15, 1: lanes 16-31
    const int l16   = lane & 15;
    const int wm    = (wave >> 2) * 64;   // wave row offset in block tile
    const int wn    = (wave & 3) * 32;    // wave col offset in block tile
    const long block_m = (long)blockIdx.y * 128;
    const long block_n = (long)blockIdx.x * 128;

    v8f acc[4][2];
#pragma unroll
    for (int i = 0; i < 4; ++i)
#pragma unroll
        for (int j = 0; j < 2; ++j)
            acc[i][j] = (v8f){0.f, 0.f, 0.f, 0.f, 0.f, 0.f, 0.f, 0.f};

    // stage one 128x32 A tile + 32x128 B tile into buffer `sb`
    auto stage = [&](int k0, int sb) {
#pragma unroll
        for (int i = 0; i < 4; ++i) {          // A: 1024 x 16B, 4 per thread
            int idx = tid + (i << 8);
            int r   = idx >> 3;
            int c4  = (idx & 7) << 2;
            cp_async16(A + (block_m + r) * (long)K + k0 + c4, &As[sb][r][c4]);
        }
#pragma unroll
        for (int i = 0; i < 4; ++i) {          // B: 1024 x 16B, guard ragged N edge
            int idx = tid + (i << 8);
            int r   = idx >> 5;
            int c4  = (idx & 31) << 2;
            long gn = block_n + c4;
            const float* src = Bg + (long)(k0 + r) * N + gn;
            float* dst = &Bs[sb][r][c4];
            if (gn + 3 < (long)N) {
                cp_async16(src, dst);
            } else {
                v4f v = (v4f){0.f, 0.f, 0.f, 0.f};
#pragma unroll
                for (int c = 0; c < 4; ++c)
                    if (gn + c < (long)N) v[c] = src[c];
                *(v4f*)dst = v;
            }
        }
    };

    const int nk = K >> 5;
    stage(0, 0);
    wait_async_lds();
    __syncthreads();

    int buf = 0;
    for (int kt = 0; kt < nk; ++kt) {
        if (kt + 1 < nk) stage((kt + 1) << 5, buf ^ 1);   // prefetch next tile

        // ---- 8 K-micro-steps of 4 on current buffer ----
#pragma unroll
        for (int kk = 0; kk < 32; kk += 4) {
            v2f af[4], bf[2];
            // A frag 16x4: lanes 0-15 hold K=kk,kk+1; lanes 16-31 hold K=kk+2,kk+3
#pragma unroll
            for (int i = 0; i < 4; ++i) {
                const float* p = &As[buf][wm + i * 16 + l16][kk + lhalf * 2];
                af[i][0] = p[0];
                af[i][1] = p[1];
            }
            // B frag 4x16: mirrored layout over N
#pragma unroll
            for (int j = 0; j < 2; ++j) {
                int col  = wn + j * 16 + l16;
                bf[j][0] = Bs[buf][kk + lhalf * 2 + 0][col];
                bf[j][1] = Bs[buf][kk + lhalf * 2 + 1][col];
            }
#pragma unroll
            for (int i = 0; i < 4; ++i)
#pragma unroll
                for (int j = 0; j < 2; ++j)
                    acc[i][j] = __builtin_amdgcn_wmma_f32_16x16x4_f32(
                        false, af[i], false, bf[j], (short)0, acc[i][j],
                        false, false);
        }

        wait_async_lds();
        __syncthreads();
        buf ^= 1;
    }

    // ---- store D: VGPR r -> row r (lanes 0-15) / r+8 (lanes 16-31), col = lane&15
#pragma unroll
    for (int i = 0; i < 4; ++i)
#pragma unroll
        for (int j = 0; j < 2; ++j) {
            long col = block_n + wn + j * 16 + l16;
            if (col < N) {
#pragma unroll
                for (int r = 0; r < 8; ++r) {
                    long row = block_m + wm + i * 16 + lhalf * 8 + r;
                    C[row * (long)N + col] = acc[i][j][r];
                }
            }
        }
}

// ---------------------------------------------------------------------------
// Depthwise causal conv1d (K=4) + SiLU over the hBC slice of proj.
// proj: [B,S,PROJ]; hbc out: [B,S,CONV_DIM]. One thread per (b,t,c), c fastest.
// ---------------------------------------------------------------------------
__global__ __launch_bounds__(256) void conv_silu_kernel(
    const float* __restrict__ proj, const float* __restrict__ w,
    const float* __restrict__ bias, float* __restrict__ hbc)
{
    long idx = (long)blockIdx.x * 256 + threadIdx.x;
    const long total = (long)BS_ * CONV_DIM_;
    if (idx >= total) return;
    int  c  = (int)(idx % CONV_DIM_);
    long bt = idx / CONV_DIM_;
    int  t  = (int)(bt % S_);
    long b  = bt / S_;

    float a = bias[c];
    const float* wk  = w + c * KCONV_;
    const float* col = proj + (b * S_) * (long)PROJ_ + INTER_ + c;
#pragma unroll
    for (int k = 0; k < KCONV_; ++k) {
        int ts = t - (KCONV_ - 1) + k;
        if (ts >= 0) a += wk[k] * col[(long)ts * PROJ_];
    }
    hbc[idx] = a / (1.f + expf(-a));   // SiLU
}

// ---------------------------------------------------------------------------
// Selective scan. One block per (b,h). State [HDIM=64][NSTATE=64] in registers:
// thread t owns d = t>>2, n in [ (t&3)*16, (t&3)*16+16 ). Double-buffered LDS
// staging (async global->LDS), one barrier per step, 4-lane shuffle reduce.
// ---------------------------------------------------------------------------
__global__ __launch_bounds__(256) void scan_kernel(
    const float* __restrict__ hbc, const float* __restrict__ proj,
    const float* __restrict__ dt_bias, const float* __restrict__ A_log,
    const float* __restrict__ Dp, float* __restrict__ y)
{
    const int bh = blockIdx.x;
    const int b  = bh >> 6;
    const int h  = bh & 63;
    const int tid = threadIdx.x;
    const int d   = tid >> 2;
    const int nq  = tid & 3;
    const int nb  = nq << 4;

    const float Ah  = -expf(A_log[h]);
    const float dtb = dt_bias[h];
    const float Dv  = Dp[h];

    float st[16];
#pragma unroll
    for (int j = 0; j < 16; ++j) st[j] = 0.f;

    __shared__ float sB[2][64], sC[2][64], sH[2][64];

    const float* hb  = hbc  + (long)b * S_ * CONV_DIM_;
    const float* dtp = proj + (long)b * S_ * PROJ_ + DT_OFF_ + h;
    float*       yb  = y    + (long)b * S_ * INTER_ + h * 64 + d;

    for (int t = 0; t < S_; ++t) {
        const int p = t & 1;
        const float* row = hb + (long)t * CONV_DIM_;
        if (tid < 64)       cp_async4(row + h * 64 + tid,              &sH[p][tid]);
        else if (tid < 128) cp_async4(row + INTER_ + (tid - 64),       &sB[p][tid - 64]);
        else if (tid < 192) cp_async4(row + INTER_ + 64 + (tid - 128), &sC[p][tid - 128]);
        wait_async_lds();
        __syncthreads();

        float dtr = dtp[(long)t * PROJ_] + dtb;
        float dt  = (dtr > 20.f) ? dtr : log1pf(expf(dtr));   // softplus
        float dA  = expf(dt * Ah);
        float hd  = sH[p][d];
        float xdt = hd * dt;

        float a = 0.f;
#pragma unroll
        for (int j = 0; j < 16; ++j) {
            float s2 = st[j] * dA + xdt * sB[p][nb + j];
            st[j] = s2;
            a += s2 * sC[p][nb + j];
        }
        a += __shfl_xor(a, 1, 32);
        a += __shfl_xor(a, 2, 32);
        if (nq == 0) yb[(long)t * INTER_] = a + Dv * hd;
        // next iteration stages the other buffer; its barrier provides safety
    }
}

// ---------------------------------------------------------------------------
// Gated RMSNorm, in-place on y. gate = first INTER cols of proj.
// g = y * silu(gate); y = norm_w * g * rsqrt(mean(g^2) + eps). One block/row.
// ---------------------------------------------------------------------------
__global__ __launch_bounds__(256) void norm_kernel(
    float* __restrict__ y, const float* __restrict__ proj,
    const float* __restrict__ nw)
{
    const long row = blockIdx.x;
    float*       yr = y + row * INTER_;
    const float* gr = proj + row * (long)PROJ_;
    const int tid = threadIdx.x;

    float g[16];
    float part = 0.f;
#pragma unroll
    for (int i = 0; i < 16; ++i) {
        int idx  = tid + (i << 8);
        float gt = gr[idx];
        float gv = yr[idx] * (gt / (1.f + expf(-gt)));
        g[i] = gv;
        part += gv * gv;
    }
    part += __shfl_xor(part, 1, 32);
    part += __shfl_xor(part, 2, 32);
    part += __shfl_xor(part, 4, 32);
    part += __shfl_xor(part, 8, 32);
    part += __shfl_xor(part, 16, 32);

    __shared__ float red[8];
    if ((tid & 31) == 0) red[tid >> 5] = part;
    __syncthreads();
    float tot = 0.f;
#pragma unroll
    for (int wv = 0; wv < 8; ++wv) tot += red[wv];
    float scale = rsqrtf(tot * (1.f / INTER_) + EPS_);

#pragma unroll
    for (int i = 0; i < 16; ++i) {
        int idx = tid + (i << 8);
        yr[idx] = nw[idx] * g[i] * scale;
    }
}

// ---------------------------------------------------------------------------
extern "C" void kernel_launch(void* const* d_in, const int* in_sizes, int n_in,
                              void* d_out, int out_size, void* d_ws, size_t ws_size,
                              hipStream_t stream)
{
    const float* x       = (const float*)d_in[0];  // [B,S,HID]
    const float* w_in    = (const float*)d_in[1];  // [HID,PROJ]
    const float* conv_w  = (const float*)d_in[2];  // [CONV_DIM,1,K]
    const float* conv_b  = (const float*)d_in[3];  // [CONV_DIM]
    const float* dt_bias = (const float*)d_in[4];  // [NHEADS]
    const float* A_log   = (const float*)d_in[5];  // [NHEADS]
    const float* Dp      = (const float*)d_in[6];  // [NHEADS]
    const float* norm_w  = (const float*)d_in[7];  // [INTER]
    const float* w_out   = (const float*)d_in[8];  // [INTER,HID]
    float* out = (float*)d_out;                    // [B,S,HID]

    // workspace layout (fp32): proj | hbc | y  (~131 MB total)
    float* proj = (float*)d_ws;                            // [BS, PROJ]
    float* hbc  = proj + (size_t)BS_ * PROJ_;              // [BS, CONV_DIM]
    float* ybuf = hbc  + (size_t)BS_ * CONV_DIM_;          // [BS, INTER]

    dim3 blk(256);

    // 1) in_proj: [2048x2048] x [2048x8384]
    gemm_f32_wmma<<<dim3((PROJ_ + 127) / 128, BS_ / 128), blk, 0, stream>>>(
        x, w_in, proj, BS_, PROJ_, HID_);

    // 2) depthwise conv + SiLU
    long convN = (long)BS_ * CONV_DIM_;
    conv_silu_kernel<<<dim3((unsigned)((convN + 255) / 256)), blk, 0, stream>>>(
        proj, conv_w, conv_b, hbc);

    // 3) selective scan (one block per (b,h))
    scan_kernel<<<dim3(B_ * NHEADS_), blk, 0, stream>>>(
        hbc, proj, dt_bias, A_log, Dp, ybuf);

    // 4) gated RMSNorm in place
    norm_kernel<<<dim3(BS_), blk, 0, stream>>>(ybuf, proj, norm_w);

    // 5) out_proj: [2048x4096] x [4096x2048]
    gemm_f32_wmma<<<dim3(HID_ / 128, BS_ / 128), blk, 0, stream>>>(
        ybuf, w_out, out, BS_, HID_, INTER_);
}